// GNN_33019708571669
// MI455X (gfx1250) — compile-verified
//
#include <hip/hip_runtime.h>
#include <hip/hip_bf16.h>

typedef float v2f __attribute__((ext_vector_type(2)));
typedef float v8f __attribute__((ext_vector_type(8)));

// ---------------------------------------------------------------------------
// Utility kernels
// ---------------------------------------------------------------------------
__global__ void zero4_k(float4* __restrict__ p, long n4) {
  long i = (long)blockIdx.x * blockDim.x + threadIdx.x;
  long st = (long)gridDim.x * blockDim.x;
  float4 z; z.x = z.y = z.z = z.w = 0.0f;
  for (; i < n4; i += st) p[i] = z;
}

__global__ void copy4_k(float4* __restrict__ dst, const float4* __restrict__ src, long n4) {
  long i = (long)blockIdx.x * blockDim.x + threadIdx.x;
  long st = (long)gridDim.x * blockDim.x;
  for (; i < n4; i += st) dst[i] = src[i];
}

// ---------------------------------------------------------------------------
// Degree + normalization
// ---------------------------------------------------------------------------
__global__ void degree_k(const int* __restrict__ dstIdx, float* __restrict__ deg, int E) {
  int e = blockIdx.x * blockDim.x + threadIdx.x;
  if (e < E) atomicAdd(&deg[dstIdx[e]], 1.0f);
}

__global__ void norm_k(const float* __restrict__ deg, float* __restrict__ dinv,
                       float* __restrict__ rdeg, int n) {
  int i = blockIdx.x * blockDim.x + threadIdx.x;
  if (i >= n) return;
  float d = deg[i];
  dinv[i] = (d > 0.0f) ? rsqrtf(d) : 0.0f;   // deg>0 => deg>=1, max() redundant
  rdeg[i] = 1.0f / fmaxf(d, 1.0f);
}

// ---------------------------------------------------------------------------
// Edge scatter: dst_feat[dst] += w * src_feat[src]   (w = dinv[d]*dinv[s] or 1)
// 16 lanes cooperate per edge, float4 per lane (64 floats per row).
// Feature tables (25.6MB) are L2-resident on MI455X (192MB L2) -> atomics hit L2.
// ---------------------------------------------------------------------------
template <bool WEIGHTED>
__global__ void scatter_k(const float* __restrict__ src_feat, float* __restrict__ dst_feat,
                          const float* __restrict__ dinv,
                          const int* __restrict__ dstIdx, const int* __restrict__ srcIdx,
                          int E) {
  long t = (long)blockIdx.x * blockDim.x + threadIdx.x;
  int e = (int)(t >> 4);
  int sub = (int)(t & 15);
  if (e >= E) return;
  int d = dstIdx[e];
  int s = srcIdx[e];
  float w = 1.0f;
  if (WEIGHTED) w = dinv[d] * dinv[s];
  const float4 v = *(const float4*)(src_feat + (long)s * 64 + sub * 4);
  float* o = dst_feat + (long)d * 64 + sub * 4;
  atomicAdd(o + 0, v.x * w);
  atomicAdd(o + 1, v.y * w);
  atomicAdd(o + 2, v.z * w);
  atomicAdd(o + 3, v.w * w);
}

// ---------------------------------------------------------------------------
// Dual-GEMM + activation, one 16x16 output tile per wave32 using
// V_WMMA_F32_16X16X4_F32 (exact fp32, K=64 -> 16 chained k-steps per matrix).
//   out = act( (agg .* rdeg) @ Wl + h @ Wr + bias )
// ACT: 0 = SELU, 1 = row-softmax over NOUT(=16) columns.
// A layout (16x4 f32): lane = M (lane&15), half-wave selects K pair, 2 VGPRs.
// C/D layout: VGPR v <-> row v + 8*(lane>>4); column = lane&15.
// Requires N % 16 == 0 (N = 100000 = 6250*16).
// ---------------------------------------------------------------------------
template <int NOUT, int ACT>
__global__ void __launch_bounds__(256)
gemm_dual_k(const float* __restrict__ agg, const float* __restrict__ rdeg,
            const float* __restrict__ h,
            const float* __restrict__ Wl, const float* __restrict__ Wr,
            const float* __restrict__ bias, float* __restrict__ out, int nTiles) {
  const int lane = threadIdx.x & 31;
  const int wave = threadIdx.x >> 5;
  constexpr int CT = NOUT / 16;           // column tiles per row block
  const int tile = blockIdx.x * 8 + wave; // 8 waves per 256-thread block
  if (tile >= nTiles) return;             // wave-uniform: EXEC stays all-ones
  const int rowTile = tile / CT;
  const int colTile = tile - rowTile * CT;
  const int m    = lane & 15;             // A row index / B,D column index
  const int half = lane >> 4;             // 0 or 1
  const int row  = rowTile * 16 + m;      // A-matrix row this lane feeds
  const int col  = colTile * 16 + m;      // output column this lane owns
  const int kh   = half * 2;              // K sub-offset within a k-step

  const float rd = rdeg[row];
  const float* aab = agg + (long)row * 64 + kh;
  const float* ahb = h   + (long)row * 64 + kh;
  const float* wl  = Wl + (long)kh * NOUT + col;
  const float* wr  = Wr + (long)kh * NOUT + col;

  v8f acc = {0.f, 0.f, 0.f, 0.f, 0.f, 0.f, 0.f, 0.f};
#pragma unroll
  for (int k0 = 0; k0 < 64; k0 += 4) {
    v2f a, b;
    a.x = aab[k0] * rd;
    a.y = aab[k0 + 1] * rd;
    b.x = wl[(long)k0 * NOUT];
    b.y = wl[(long)(k0 + 1) * NOUT];
    acc = __builtin_amdgcn_wmma_f32_16x16x4_f32(false, a, false, b, (short)0, acc,
                                                false, false);
  }
#pragma unroll
  for (int k0 = 0; k0 < 64; k0 += 4) {
    v2f a, b;
    a.x = ahb[k0];
    a.y = ahb[k0 + 1];
    b.x = wr[(long)k0 * NOUT];
    b.y = wr[(long)(k0 + 1) * NOUT];
    acc = __builtin_amdgcn_wmma_f32_16x16x4_f32(false, a, false, b, (short)0, acc,
                                                false, false);
  }

  const float bv = bias[col];
  float r[8];
#pragma unroll
  for (int v = 0; v < 8; ++v) r[v] = acc[v] + bv;

  if (ACT == 0) {
    const float kScale = 1.0507009873554805f;
    const float kAlpha = 1.6732632423543772f;
#pragma unroll
    for (int v = 0; v < 8; ++v)
      r[v] = (r[v] > 0.0f) ? kScale * r[v] : kScale * kAlpha * (expf(r[v]) - 1.0f);
  } else {
    // One output row spans the 16 lanes of a half-wave in a single VGPR;
    // xor-shuffles with masks 1/2/4/8 never cross the half-wave boundary.
#pragma unroll
    for (int v = 0; v < 8; ++v) {
      float mx = r[v];
#pragma unroll
      for (int dlt = 1; dlt < 16; dlt <<= 1) mx = fmaxf(mx, __shfl_xor(mx, dlt, 32));
      float e = expf(r[v] - mx);
      float s = e;
#pragma unroll
      for (int dlt = 1; dlt < 16; dlt <<= 1) s += __shfl_xor(s, dlt, 32);
      r[v] = e / s;
    }
  }

  const int outRow0 = rowTile * 16 + half * 8;
  const int outCol  = colTile * 16 + m;
#pragma unroll
  for (int v = 0; v < 8; ++v)
    out[(long)(outRow0 + v) * NOUT + outCol] = r[v];
}

// ---------------------------------------------------------------------------
// Orchestration
// ---------------------------------------------------------------------------
extern "C" void kernel_launch(void* const* d_in, const int* in_sizes, int n_in,
                              void* d_out, int out_size, void* d_ws, size_t ws_size,
                              hipStream_t stream) {
  const float* x   = (const float*)d_in[0];
  const int*   ei  = (const int*)d_in[1];
  const float* W2l = (const float*)d_in[2];
  const float* W2r = (const float*)d_in[3];
  const float* b2  = (const float*)d_in[4];
  const float* W3l = (const float*)d_in[5];
  const float* W3r = (const float*)d_in[6];
  const float* b3  = (const float*)d_in[7];

  const int N = in_sizes[0] / 64;
  const int E = in_sizes[1] / 2;
  const int* dstIdx = ei;        // edge_index[0] = dst
  const int* srcIdx = ei + E;    // edge_index[1] = src

  // Workspace carve-up (256B aligned slices)
  char*  ws  = (char*)d_ws;
  size_t off = 0;
  auto take = [&](size_t bytes) -> char* {
    char* p = ws + off;
    off += (bytes + 255) & ~(size_t)255;
    return p;
  };
  float* deg  = (float*)take((size_t)N * sizeof(float));
  float* dinv = (float*)take((size_t)N * sizeof(float));
  float* rdeg = (float*)take((size_t)N * sizeof(float));
  float* hA   = (float*)take((size_t)N * 64 * sizeof(float));
  float* hB   = (float*)take((size_t)N * 64 * sizeof(float));
  float* hC   = (float*)take((size_t)N * 64 * sizeof(float));

  const long NF = (long)N * 64;
  const int  ZG = 4096;  // grid-stride blocks for zero/copy

  const int nodeBlocks    = (N + 255) / 256;
  const int edgeBlocks    = (E + 255) / 256;
  const long scatThreads  = (long)E * 16;
  const int  scatBlocks   = (int)((scatThreads + 255) / 256);
  const int  rowTiles     = N / 16;                 // N = 100000 -> 6250

  // 1) degree + gcn_norm factors
  zero4_k<<<ZG, 256, 0, stream>>>((float4*)deg, N / 4);
  degree_k<<<edgeBlocks, 256, 0, stream>>>(dstIdx, deg, E);
  norm_k<<<nodeBlocks, 256, 0, stream>>>(deg, dinv, rdeg, N);

  // 2) KProp step 1: hA = x + A_norm @ x
  copy4_k<<<ZG, 256, 0, stream>>>((float4*)hA, (const float4*)x, NF / 4);
  scatter_k<true><<<scatBlocks, 256, 0, stream>>>(x, hA, dinv, dstIdx, srcIdx, E);

  //    KProp step 2: hB = hA + A_norm @ hA
  copy4_k<<<ZG, 256, 0, stream>>>((float4*)hB, (const float4*)hA, NF / 4);
  scatter_k<true><<<scatBlocks, 256, 0, stream>>>(hA, hB, dinv, dstIdx, srcIdx, E);

  // 3) SAGE layer 1: hC = sum_nbr(hB); hA = selu(hC/deg @ W2l + hB @ W2r + b2)
  zero4_k<<<ZG, 256, 0, stream>>>((float4*)hC, NF / 4);
  scatter_k<false><<<scatBlocks, 256, 0, stream>>>(hB, hC, nullptr, dstIdx, srcIdx, E);
  {
    const int nTiles = rowTiles * (64 / 16);
    gemm_dual_k<64, 0><<<(nTiles + 7) / 8, 256, 0, stream>>>(hC, rdeg, hB, W2l, W2r,
                                                             b2, hA, nTiles);
  }

  // 4) SAGE layer 2 + softmax: hC = sum_nbr(hA); out = softmax(hC/deg@W3l + hA@W3r + b3)
  zero4_k<<<ZG, 256, 0, stream>>>((float4*)hC, NF / 4);
  scatter_k<false><<<scatBlocks, 256, 0, stream>>>(hA, hC, nullptr, dstIdx, srcIdx, E);
  {
    const int nTiles = rowTiles;  // 16 output cols -> single column tile
    gemm_dual_k<16, 1><<<(nTiles + 7) / 8, 256, 0, stream>>>(hC, rdeg, hA, W3l, W3r,
                                                             b3, (float*)d_out, nTiles);
  }
}